// RPNHead_52518860095513
// MI455X (gfx1250) — compile-verified
//
#include <hip/hip_runtime.h>
#include <hip/hip_bf16.h>

// RPN head (conv3x3+relu+whiten -> FALKON gaussian scores -> box regression)
// for MI455X (gfx1250, wave32). All GEMMs run on V_WMMA_F32_16X16X4_F32.
// FALKON tile staging uses GLOBAL_LOAD_ASYNC_TO_LDS_B128 (ASYNCcnt).
//
// Workspace layout (floats), total ~45.6 MB:
//   wP   [9][512][1024][2] : conv weights as [tap][ci/2][co][ci%2]
//   F    [1904][1024]      : whitened features, rows padded 1900->1904
//   fsq  [1904]            : ||f_a||^2
//   nysq [15][1008]        : ||ny_{c,m}||^2, m padded 1000->1008 (pad = 0)

#define HH 38
#define WW 50
#define CC 1024
#define AREA (HH*WW)          // 1900
#define APAD 1904             // 119 * 16
#define NCLS 15
#define MM 1000
#define MPAD 1008             // 63 * 16
#define NTILES 63
#define ATILES 119

typedef float v2f __attribute__((ext_vector_type(2)));
typedef float v8f __attribute__((ext_vector_type(8)));

__device__ __forceinline__ v8f wmma_f32(v2f a, v2f b, v8f c) {
  // (neg_a, A, neg_b, B, c_mod, C, reuse_a, reuse_b)
  return __builtin_amdgcn_wmma_f32_16x16x4_f32(false, a, false, b, (short)0, c,
                                               false, false);
}

// Async DMA: global -> LDS, 16B per lane, tracked by ASYNCcnt.
// lds_addr: 32-bit LDS byte address (low 32 bits of the generic pointer,
// as mapped by the LDS aperture).
__device__ __forceinline__ void async_ld_b128(const void* lds_ptr,
                                              unsigned voffset,
                                              const void* sbase) {
  unsigned lds_addr = (unsigned)(uintptr_t)lds_ptr;
  asm volatile("global_load_async_to_lds_b128 %0, %1, %2 offset:0"
               :: "v"(lds_addr), "v"(voffset), "s"(sbase)
               : "memory");
}
__device__ __forceinline__ void wait_async0() {
  asm volatile("s_wait_asynccnt 0" ::: "memory");
}

// ---------------------------------------------------------------------------
// Kernel 0: repack conv weights [co][ci][kh][kw] -> [tap][ci/2][co][ci%2]
// so conv B-fragments are single coalesced float2 loads.
// ---------------------------------------------------------------------------
__global__ __launch_bounds__(256) void wtrans_kernel(const float* __restrict__ w,
                                                     float* __restrict__ wP) {
  const int idx = blockIdx.x;          // 0 .. 9*512
  const int ci2 = idx & 511;
  const int tap = idx >> 9;            // 0..8
  const int kh  = tap / 3, kw = tap % 3;
  for (int co = threadIdx.x; co < CC; co += 256) {
    float2 v;
    v.x = w[(((size_t)co * CC + ci2 * 2 + 0) * 3 + kh) * 3 + kw];
    v.y = w[(((size_t)co * CC + ci2 * 2 + 1) * 3 + kh) * 3 + kw];
    *reinterpret_cast<float2*>(
        &wP[(((size_t)tap * 512 + ci2) * CC + co) * 2]) = v;
  }
}

// ---------------------------------------------------------------------------
// Kernel 1: conv3x3 + bias + relu + whitening -> F [APAD][CC]
// grid (38, 32), block 256 (8 waves).
// wave w: position tile pt = w&3 (w-cols pt*16..), channel tile ct = w>>2.
// x patch in LDS as [dh][col][ci] (K innermost, stride 12) so every WMMA
// A-fragment is one aligned ds_load_b64 straight into a register pair.
// ---------------------------------------------------------------------------
__global__ __launch_bounds__(256) void conv_kernel(
    const float* __restrict__ x, const float* __restrict__ wP,
    const float* __restrict__ conv_b, const float* __restrict__ stats_mean,
    const float* __restrict__ mean_norm, float* __restrict__ F) {
  __shared__ float XS[3][68][12];  // [dh][col][ci-in-chunk], col 0 <-> ww=-1

  const int tid  = threadIdx.x;
  const int lane = tid & 31;
  const int wv   = tid >> 5;
  const int h    = blockIdx.x;
  const int cg   = blockIdx.y;
  const int pt   = wv & 3;
  const int ct   = wv >> 2;
  const int nl   = lane & 15;
  const int co   = cg * 32 + ct * 16 + nl;
  const int klo  = (lane < 16) ? 0 : 2;

  v8f acc = {0.f, 0.f, 0.f, 0.f, 0.f, 0.f, 0.f, 0.f};

  for (int ci0 = 0; ci0 < CC; ci0 += 8) {
    if (tid == 0 && ci0 + 8 < CC)  // global_prefetch_b8 for the next chunk
      __builtin_prefetch(&x[((size_t)(ci0 + 8) * HH + h) * WW], 0, 1);

    __syncthreads();
    for (int i = tid; i < 8 * 3 * 68; i += 256) {
      const int cc   = i % 68;       // coalesced over col
      const int rem  = i / 68;
      const int dh   = rem % 3;
      const int cidx = rem / 3;
      const int ww2  = cc - 1;
      const int hh2  = h + dh - 1;
      float v = 0.f;
      if (hh2 >= 0 && hh2 < HH && ww2 >= 0 && ww2 < WW)
        v = x[((size_t)(ci0 + cidx) * HH + hh2) * WW + ww2];
      XS[dh][cc][cidx] = v;
    }
    __syncthreads();

#pragma unroll
    for (int kg = 0; kg < 2; ++kg) {
#pragma unroll
      for (int tap = 0; tap < 9; ++tap) {
        const int dh = tap / 3, dw = tap % 3;
        const int kk = kg * 4 + klo;              // even
        const int col = pt * 16 + nl + dw;
        const float2 apair = *reinterpret_cast<const float2*>(&XS[dh][col][kk]);
        v2f a, b;
        a.x = apair.x;
        a.y = apair.y;
        const int kci = ci0 + kk;                 // even
        const float2 wpair = *reinterpret_cast<const float2*>(
            &wP[(((size_t)tap * 512 + (kci >> 1)) * CC + co) * 2]);
        b.x = wpair.x;
        b.y = wpair.y;
        acc = wmma_f32(a, b, acc);
      }
    }
  }

  // epilogue: bias + relu + whitening, write F
  const float bn = conv_b[co];
  const float mn = stats_mean[co];
  const float scale = 20.f / mean_norm[0];
#pragma unroll
  for (int v = 0; v < 8; ++v) {
    const int m  = (lane < 16) ? v : v + 8;
    const int wq = pt * 16 + m;
    if (wq < WW) {
      float t = acc[v] + bn;
      t = fmaxf(t, 0.f);
      F[(size_t)(h * WW + wq) * CC + co] = (t - mn) * scale;
    }
  }
}

// ---------------------------------------------------------------------------
// Kernel 2: row norms of F
// ---------------------------------------------------------------------------
__global__ __launch_bounds__(32) void fsq_kernel(const float* __restrict__ F,
                                                 float* __restrict__ fsq) {
  const int row = blockIdx.x;
  const int lane = threadIdx.x;
  float s = 0.f;
  for (int k = lane; k < CC; k += 32) {
    const float t = F[(size_t)row * CC + k];
    s += t * t;
  }
  for (int off = 16; off; off >>= 1) s += __shfl_xor(s, off, 32);
  if (lane == 0) fsq[row] = s;
}

// ---------------------------------------------------------------------------
// Kernel 3: nystrom row norms, padded to MPAD (pad entries = 0)
// grid (1008, 15), block 32
// ---------------------------------------------------------------------------
__global__ __launch_bounds__(32) void nysq_kernel(const float* __restrict__ ny,
                                                  float* __restrict__ nysq) {
  const int n = blockIdx.x;
  const int c = blockIdx.y;
  const int lane = threadIdx.x;
  float s = 0.f;
  if (n < MM) {
    for (int k = lane; k < CC; k += 32) {
      const float t = ny[((size_t)c * MM + n) * CC + k];
      s += t * t;
    }
    for (int off = 16; off; off >>= 1) s += __shfl_xor(s, off, 32);
  }
  if (lane == 0) nysq[c * MPAD + n] = s;
}

// ---------------------------------------------------------------------------
// Kernel 4: FALKON scores. grid (119, 15), block 256 (8 waves).
// Each wave accumulates 8 n-tiles (persistent v8f acc each) over K chunks of
// 64. A tile async-DMA'd to LDS (shared by all waves), B tiles async-DMA'd
// per wave (sync fallback for the 1000->1008 pad tile). A fragments cached in
// VGPRs across the 8 n-tile groups.
// Fused: d2 -> exp -> * alpha -> cross-lane reduce -> obj output.
// ---------------------------------------------------------------------------
__global__ __launch_bounds__(256) void falkon_kernel(
    const float* __restrict__ F, const float* __restrict__ fsq,
    const float* __restrict__ nysq, const float* __restrict__ ny,
    const float* __restrict__ alpha, const int* __restrict__ vmask,
    float* __restrict__ out) {
  __shared__ float As[16][68];       // [m][kk], padded vs bank conflicts
  __shared__ float Bs[8][16][68];    // per-wave [n][kk]
  __shared__ float fsqs[16];
  __shared__ float scw[8][16];       // per-wave score partials (deterministic)

  const int tid  = threadIdx.x;
  const int lane = tid & 31;
  const int wv   = tid >> 5;
  const int a0   = blockIdx.x * 16;
  const int c    = blockIdx.y;
  const int nl   = lane & 15;
  const int klo  = (lane < 16) ? 0 : 2;
  const float* nyc = ny + (size_t)c * MM * CC;   // uniform class base

  if (tid < 16) fsqs[tid] = fsq[a0 + tid];
  if (tid < 128) scw[tid >> 4][tid & 15] = 0.f;

  v8f acc[8];
#pragma unroll
  for (int g = 0; g < 8; ++g) {
    v8f z = {0.f, 0.f, 0.f, 0.f, 0.f, 0.f, 0.f, 0.f};
    acc[g] = z;
  }

  // per-thread A DMA slot: row tid>>4, 4 floats at (tid&15)*4
  const int am = tid >> 4;
  const int ak = (tid & 15) * 4;
  // per-lane B DMA slot: 2 rows per issue, 4 floats at (lane&15)*4
  const int br = lane >> 4;            // 0/1: row within pair
  const int bk = nl * 4;

  for (int kc = 0; kc < 16; ++kc) {       // K chunks of 64
    const int k0 = kc * 64;
    __syncthreads();                      // prior readers of As done
    async_ld_b128(&As[am][ak],
                  (unsigned)(((a0 + am) * CC + k0 + ak) * 4), F);
    wait_async0();
    __syncthreads();                      // As visible to all waves

    // cache this lane's A fragments for the whole chunk (reused by 8 tiles)
    float2 afrag[16];
#pragma unroll
    for (int kg = 0; kg < 16; ++kg) {
      const int kb = kg * 4 + klo;
      afrag[kg].x = As[nl][kb];
      afrag[kg].y = As[nl][kb + 1];
    }

    for (int grp = 0; grp < 8; ++grp) {
      const int nt = grp * 8 + wv;
      if (nt >= NTILES) continue;
      if (nt == NTILES - 1) {
        // boundary tile: rows >= MM must be zero -> synchronous guarded path
        for (int r = 0; r < 16; ++r) {
          const int n = nt * 16 + r;
          float2 val = make_float2(0.f, 0.f);
          if (n < MM) {
            val = *reinterpret_cast<const float2*>(
                &nyc[(size_t)n * CC + k0 + lane * 2]);
          }
          Bs[wv][r][lane * 2]     = val.x;
          Bs[wv][r][lane * 2 + 1] = val.y;
        }
      } else {
        // async DMA: 8 issues x (2 rows x 16 lanes x 16B) = 16x64 floats
        const int nb = nt * 16;
#pragma unroll
        for (int rp = 0; rp < 16; rp += 2) {
          const int n = nb + rp + br;
          async_ld_b128(&Bs[wv][rp + br][bk],
                        (unsigned)((n * CC + k0 + bk) * 4), nyc);
        }
        wait_async0();
      }
#pragma unroll
      for (int kg = 0; kg < 16; ++kg) {
        const int kb = kg * 4 + klo;
        v2f a, b;
        a.x = afrag[kg].x;
        a.y = afrag[kg].y;
        b.x = Bs[wv][nl][kb];
        b.y = Bs[wv][nl][kb + 1];
        acc[grp] = wmma_f32(a, b, acc[grp]);
      }
    }
  }

  // d2 -> K -> alpha-weighted reduce over n within each tile
  for (int grp = 0; grp < 8; ++grp) {
    const int nt = grp * 8 + wv;
    if (nt >= NTILES) continue;
    const int n = nt * 16 + nl;
    const float nys = nysq[c * MPAD + n];
    const float al  = (n < MM) ? alpha[(size_t)c * MM + n] : 0.f;
#pragma unroll
    for (int v = 0; v < 8; ++v) {
      const int mloc = (lane < 16) ? v : v + 8;
      const float d2 = fmaxf(fsqs[mloc] + nys - 2.f * acc[grp][v], 0.f);
      float s = __expf(d2 * (-1.0f / 450.0f)) * al;
      for (int off = 1; off < 16; off <<= 1) s += __shfl_xor(s, off, 16);
      if ((lane & 15) == 0) scw[wv][mloc] += s;   // wave-exclusive slot
    }
  }
  __syncthreads();

  if (tid < 16) {
    float sum = 0.f;
#pragma unroll
    for (int w = 0; w < 8; ++w) sum += scw[w][tid];
    const int a = a0 + tid;
    if (a < AREA) {
      const float mc = (float)vmask[c];
      out[(size_t)c * AREA + a] = sum * mc - 2.f * (1.f - mc);
    }
  }
}

// ---------------------------------------------------------------------------
// Kernel 5: box regression. grid (119, 15), block 32 (one wave).
// Y = F @ beta[:-1] + beta[-1]; Y = Y @ T_inv + mu; Y *= mask.
// ---------------------------------------------------------------------------
__global__ __launch_bounds__(32) void bbox_kernel(
    const float* __restrict__ F, const float* __restrict__ beta,
    const float* __restrict__ Tinv, const float* __restrict__ mu,
    const int* __restrict__ vmask, float* __restrict__ out) {
  __shared__ float Yt[16][4];

  const int lane = threadIdx.x;
  const int a0   = blockIdx.x * 16;
  const int c    = blockIdx.y;
  const int nl   = lane & 15;
  const int klo  = (lane < 16) ? 0 : 2;

  v8f acc = {0.f, 0.f, 0.f, 0.f, 0.f, 0.f, 0.f, 0.f};
  for (int kg = 0; kg < 256; ++kg) {
    const int kb = kg * 4 + klo;
    v2f a, b;
    a.x = F[(size_t)(a0 + nl) * CC + kb];
    a.y = F[(size_t)(a0 + nl) * CC + kb + 1];
    if (nl < 4) {
      b.x = beta[((size_t)c * (CC + 1) + kb) * 4 + nl];
      b.y = beta[((size_t)c * (CC + 1) + kb + 1) * 4 + nl];
    } else {
      b.x = 0.f;
      b.y = 0.f;
    }
    acc = wmma_f32(a, b, acc);
  }

  if (nl < 4) {
    const float bias = beta[((size_t)c * (CC + 1) + CC) * 4 + nl];
#pragma unroll
    for (int v = 0; v < 8; ++v) {
      const int m = (lane < 16) ? v : v + 8;
      Yt[m][nl] = acc[v] + bias;
    }
  }
  __syncthreads();

  if (lane < 16) {
    const int m = lane;
    const int a = a0 + m;
    if (a < AREA) {
      const float mc = (float)vmask[c];
#pragma unroll
      for (int j2 = 0; j2 < 4; ++j2) {
        float y = mu[c * 4 + j2];
#pragma unroll
        for (int j = 0; j < 4; ++j) y += Yt[m][j] * Tinv[(c * 4 + j) * 4 + j2];
        out[(size_t)(NCLS + c * 4 + j2) * AREA + a] = y * mc;
      }
    }
  }
}

// ---------------------------------------------------------------------------
extern "C" void kernel_launch(void* const* d_in, const int* in_sizes, int n_in,
                              void* d_out, int out_size, void* d_ws,
                              size_t ws_size, hipStream_t stream) {
  const float* x          = (const float*)d_in[0];
  const float* conv_w     = (const float*)d_in[1];
  const float* conv_b     = (const float*)d_in[2];
  const float* stats_mean = (const float*)d_in[3];
  const float* mean_norm  = (const float*)d_in[4];
  const float* nystrom    = (const float*)d_in[5];
  const float* alpha      = (const float*)d_in[6];
  const float* beta       = (const float*)d_in[7];
  const float* T_inv      = (const float*)d_in[8];
  const float* mu         = (const float*)d_in[9];
  const int*   vmask      = (const int*)d_in[10];
  float* out = (float*)d_out;

  float* ws   = (float*)d_ws;              // needs ~45.6 MB
  float* wP   = ws;                        // 9*1024*1024
  float* F    = wP + 9 * 1024 * 1024;      // 1904*1024
  float* fsq  = F + (size_t)APAD * CC;     // 1904
  float* nysq = fsq + APAD;                // 15*1008

  wtrans_kernel<<<dim3(9 * 512), 256, 0, stream>>>(conv_w, wP);
  conv_kernel<<<dim3(HH, CC / 32), 256, 0, stream>>>(x, wP, conv_b, stats_mean,
                                                     mean_norm, F);
  fsq_kernel<<<dim3(AREA), 32, 0, stream>>>(F, fsq);
  nysq_kernel<<<dim3(MPAD, NCLS), 32, 0, stream>>>(nystrom, nysq);
  falkon_kernel<<<dim3(ATILES, NCLS), 256, 0, stream>>>(F, fsq, nysq, nystrom,
                                                        alpha, vmask, out);
  bbox_kernel<<<dim3(ATILES, NCLS), 32, 0, stream>>>(F, beta, T_inv, mu, vmask,
                                                     out);
}